// AnchorHeadBase_32959579030073
// MI455X (gfx1250) — compile-verified
//
#include <hip/hip_runtime.h>
#include <hip/hip_bf16.h>
#include <math.h>

// ---------------- problem constants ----------------
#define BATCH   4
#define C_IN    256
#define Hh      200
#define Ww      176
#define HWp     (Hh * Ww)          // 35200 pixels per batch
#define A_ANCH  6
#define NUM_CLS 4
#define N_ANCH  (A_ANCH * HWp)     // 211200
#define TOPK    100
#define OUTP    80                 // 66 real output channels padded to 5*16
#define PITCH   17                 // LDS staging pitch (bank-conflict break)
#define NTILES  (HWp / 16)         // 2200 16-pixel tiles per batch
#define WELEMS  (C_IN * OUTP)      // 20480 f32 weight-table elements

typedef float        f32x2 __attribute__((ext_vector_type(2)));
typedef float        f32x8 __attribute__((ext_vector_type(8)));
typedef unsigned int u32x4 __attribute__((ext_vector_type(4)));
typedef int          i32x4 __attribute__((ext_vector_type(4)));
typedef int          i32x8 __attribute__((ext_vector_type(8)));

// ---------------- ws layout (bytes) ----------------
#define WS_WT     0          // Wt interleaved [128 pairs][80 out][2]  (81920 B)
#define WS_MAXSC  81920      // [B][N_ANCH] f32                        (3379200 B)
#define WS_SCORES 3461120    // [B][N_ANCH][4] f32                     (13516800 B)
#define WS_BOXES  16977920   // [B][N_ANCH][7] f32                     (23654400 B)

// =====================================================================
// Kernel 1: build channel-pair-interleaved padded weight table
//   Wt[(c>>1)*160 + o*2 + (c&1)] = W[o][c]   (o padded 66 -> 80)
// =====================================================================
__global__ __launch_bounds__(256) void prep_weights(const float* __restrict__ cls_w,
                                                    const float* __restrict__ reg_w,
                                                    float* __restrict__ Wt) {
    int i = blockIdx.x * 256 + threadIdx.x;          // over 256*80
    if (i < WELEMS) {
        int c = i / OUTP;
        int o = i % OUTP;
        float v = 0.0f;
        if (o < 24)      v = cls_w[o * C_IN + c];
        else if (o < 66) v = reg_w[(o - 24) * C_IN + c];
        Wt[(c >> 1) * (2 * OUTP) + o * 2 + (c & 1)] = v;
    }
}

// =====================================================================
// Kernel 2: fused WMMA GEMM heads + softmax + box decode + max-score
//   grid = (138, B), block = 256 (8 waves); each wave: 2x16 pixels x 80 ch
// =====================================================================
__global__ __launch_bounds__(256) void head_fused(const float* __restrict__ x,
                                                  const float* __restrict__ Wt,
                                                  const float* __restrict__ cls_b,
                                                  const float* __restrict__ reg_b,
                                                  const float* __restrict__ anchors,
                                                  float* __restrict__ maxsc,
                                                  float* __restrict__ scores,
                                                  float* __restrict__ boxes) {
    __shared__ float sW[WELEMS];                     // 80 KB, pair-interleaved
    __shared__ float sOut[8][OUTP * PITCH];          // 43.5 KB staging

    const int tid  = threadIdx.x;
    const int wave = tid >> 5;
    const int lane = tid & 31;
    const int n    = lane & 15;                      // pixel col / A-matrix row
    const int kh   = lane >> 4;                      // K half select (0/1)
    const int b    = blockIdx.y;

    int t0 = blockIdx.x * 16 + wave * 2;             // first of 2 pixel tiles
    if (t0 > NTILES - 2) t0 = NTILES - 2;            // clamp tail (dup work, same values)
    const int p0 = t0 * 16;

    // --- TDM: DMA the 80KB weight table into LDS (wave 0 issues) ---
    if (tid < 32) {
        unsigned long long ga = (unsigned long long)(size_t)Wt;   // global addr
        unsigned int       la = (unsigned int)(size_t)sW;         // LDS offset
        u32x4 g0;
        g0[0] = 1u;                                   // count=1, flags=0
        g0[1] = la;                                   // lds_addr
        g0[2] = (unsigned int)ga;                     // global_addr[31:0]
        g0[3] = (unsigned int)(ga >> 32) | 0x80000000u; // global_addr[56:32] | type=2
        i32x8 g1;
        g1[0] = 0x20000;                              // wg_mask=0, data_size=4B
        g1[1] = (int)(WELEMS << 16);                  // tensor_dim0[15:0] in [63:48]
        g1[2] = (int)((WELEMS >> 16) | (1u << 16));   // tensor_dim0 hi | tensor_dim1=1
        g1[3] = (int)(WELEMS << 16);                  // tile_dim0 = 20480
        g1[4] = 1;                                    // tile_dim1=1, tile_dim2=0
        g1[5] = WELEMS;                               // tensor_dim0_stride lo
        g1[6] = 0;
        g1[7] = 0;
        i32x4 gz  = {0, 0, 0, 0};
        i32x8 gz8 = {0, 0, 0, 0, 0, 0, 0, 0};
        __builtin_amdgcn_tensor_load_to_lds(g0, g1, gz, gz, gz8, 0);
        __builtin_amdgcn_s_wait_tensorcnt(0);
    }
    __syncthreads();

    // --- WMMA GEMM: 2 pixel tiles, 5 output tiles each ---
    f32x8 acc0[5], acc1[5];
    #pragma unroll
    for (int t = 0; t < 5; ++t) {
        acc0[t] = (f32x8){0,0,0,0,0,0,0,0};
        acc1[t] = (f32x8){0,0,0,0,0,0,0,0};
    }

    const float* Xb  = x + (size_t)b * C_IN * HWp + (size_t)(2 * kh) * HWp;
    const float* Xp0 = Xb + p0 + n;          // tile 0 pixels
    const float* Xp1 = Xb + p0 + 16 + n;     // tile 1 pixels
    const f32x2* Aw  = (const f32x2*)sW + kh * OUTP + n;  // pair-interleaved A

    for (int k = 0; k < C_IN; k += 4) {
        f32x2 bm0, bm1;
        bm0.x = Xp0[0];  bm0.y = Xp0[HWp];
        bm1.x = Xp1[0];  bm1.y = Xp1[HWp];
        Xp0 += 4 * HWp;  Xp1 += 4 * HWp;
        const f32x2* ap = Aw + (k >> 1) * OUTP;
        #pragma unroll
        for (int t = 0; t < 5; ++t) {
            f32x2 am = ap[t * 16];           // single aligned ds_load_b64
            acc0[t] = __builtin_amdgcn_wmma_f32_16x16x4_f32(
                false, am, false, bm0, (short)0, acc0[t], false, false);
            acc1[t] = __builtin_amdgcn_wmma_f32_16x16x4_f32(
                false, am, false, bm1, (short)0, acc1[t], false, false);
        }
    }

    // --- two staging+postprocess phases (tile 0, tile 1) ---
    float* so = &sOut[wave][0];
    #pragma unroll
    for (int half = 0; half < 2; ++half) {
        __syncthreads();   // previous phase readers done / GEMM done
        #pragma unroll
        for (int t = 0; t < 5; ++t) {
            #pragma unroll
            for (int r = 0; r < 8; ++r) {
                float v = half ? acc1[t][r] : acc0[t][r];
                so[(t * 16 + r + 8 * kh) * PITCH + n] = v;
            }
        }
        __syncthreads();

        const int pbase = p0 + half * 16;
        #pragma unroll
        for (int it = 0; it < 3; ++it) {
            int wi  = it * 32 + lane;        // 0..95 = 16 pixels x 6 anchors
            int pix = wi / A_ANCH;
            int a   = wi % A_ANCH;
            size_t nidx = (size_t)(pbase + pix) * A_ANCH + a;

            // softmax over 4 class logits
            float l0 = so[(a * 4 + 0) * PITCH + pix] + cls_b[a * 4 + 0];
            float l1 = so[(a * 4 + 1) * PITCH + pix] + cls_b[a * 4 + 1];
            float l2 = so[(a * 4 + 2) * PITCH + pix] + cls_b[a * 4 + 2];
            float l3 = so[(a * 4 + 3) * PITCH + pix] + cls_b[a * 4 + 3];
            float m  = fmaxf(fmaxf(l0, l1), fmaxf(l2, l3));
            float e0 = __expf(l0 - m), e1 = __expf(l1 - m);
            float e2 = __expf(l2 - m), e3 = __expf(l3 - m);
            float inv = 1.0f / (e0 + e1 + e2 + e3);
            float s0 = e0 * inv, s1 = e1 * inv, s2 = e2 * inv, s3 = e3 * inv;

            // SECOND residual box decode
            float d[7];
            #pragma unroll
            for (int j = 0; j < 7; ++j)
                d[j] = so[(24 + a * 7 + j) * PITCH + pix] + reg_b[a * 7 + j];
            const float* anc = anchors + nidx * 7;
            float xa = anc[0], ya = anc[1], za = anc[2];
            float dxa = anc[3], dya = anc[4], dza = anc[5], ra = anc[6];
            float diag = sqrtf(dxa * dxa + dya * dya);
            float bx0 = d[0] * diag + xa;
            float bx1 = d[1] * diag + ya;
            float bx2 = d[2] * dza + za;
            float bx3 = __expf(d[3]) * dxa;
            float bx4 = __expf(d[4]) * dya;
            float bx5 = __expf(d[5]) * dza;
            float bx6 = d[6] + ra;

            size_t off = (size_t)b * N_ANCH + nidx;
            maxsc[off] = fmaxf(fmaxf(s1, s2), s3);
            float* sp = scores + off * 4;
            sp[0] = s0; sp[1] = s1; sp[2] = s2; sp[3] = s3;
            float* bp = boxes + off * 7;
            bp[0] = bx0; bp[1] = bx1; bp[2] = bx2; bp[3] = bx3;
            bp[4] = bx4; bp[5] = bx5; bp[6] = bx6;
        }
    }
}

// =====================================================================
// Kernel 3: per-batch top-K via iterative block argmax (K=100)
// =====================================================================
__global__ __launch_bounds__(1024) void topk_select(float* __restrict__ maxsc,
                                                    const float* __restrict__ scores,
                                                    const float* __restrict__ boxes,
                                                    float* __restrict__ out_scores,
                                                    float* __restrict__ out_boxes) {
    __shared__ float sv[1024];
    __shared__ int   si[1024];
    const int b   = blockIdx.x;
    const int tid = threadIdx.x;
    float* ms = maxsc + (size_t)b * N_ANCH;

    for (int k = 0; k < TOPK; ++k) {
        float best = -3.0e38f;
        int   bi   = 0x7fffffff;
        for (int i = tid; i < N_ANCH; i += 1024) {
            float v = ms[i];
            if (v > best) { best = v; bi = i; }
        }
        sv[tid] = best; si[tid] = bi;
        __syncthreads();
        #pragma unroll
        for (int s = 512; s > 0; s >>= 1) {
            if (tid < s) {
                float v2 = sv[tid + s]; int i2 = si[tid + s];
                if (v2 > sv[tid] || (v2 == sv[tid] && i2 < si[tid])) {
                    sv[tid] = v2; si[tid] = i2;
                }
            }
            __syncthreads();
        }
        if (tid == 0) {
            int nsel = si[0];
            ms[nsel] = -3.0e38f;
            size_t src = (size_t)b * N_ANCH + nsel;
            float* os = out_scores + ((size_t)b * TOPK + k) * 4;
            const float* sp = scores + src * 4;
            os[0] = sp[0]; os[1] = sp[1]; os[2] = sp[2]; os[3] = sp[3];
            float* ob = out_boxes + ((size_t)b * TOPK + k) * 7;
            const float* bp = boxes + src * 7;
            #pragma unroll
            for (int j = 0; j < 7; ++j) ob[j] = bp[j];
        }
        __syncthreads();
    }
}

// =====================================================================
extern "C" void kernel_launch(void* const* d_in, const int* in_sizes, int n_in,
                              void* d_out, int out_size, void* d_ws, size_t ws_size,
                              hipStream_t stream) {
    const float* x       = (const float*)d_in[0];
    const float* cls_w   = (const float*)d_in[1];
    const float* cls_b   = (const float*)d_in[2];
    const float* reg_w   = (const float*)d_in[3];
    const float* reg_b   = (const float*)d_in[4];
    const float* anchors = (const float*)d_in[5];

    char* ws = (char*)d_ws;
    float* Wt     = (float*)(ws + WS_WT);
    float* maxsc  = (float*)(ws + WS_MAXSC);
    float* scores = (float*)(ws + WS_SCORES);
    float* boxes  = (float*)(ws + WS_BOXES);

    float* out_scores = (float*)d_out;                       // [B,100,4]
    float* out_boxes  = (float*)d_out + BATCH * TOPK * 4;    // [B,100,7]

    prep_weights<<<(WELEMS + 255) / 256, 256, 0, stream>>>(cls_w, reg_w, Wt);

    dim3 grid((NTILES + 15) / 16, BATCH);   // 138 x 4 blocks, 8 waves each
    head_fused<<<grid, 256, 0, stream>>>(x, Wt, cls_b, reg_b, anchors,
                                         maxsc, scores, boxes);

    topk_select<<<BATCH, 1024, 0, stream>>>(maxsc, scores, boxes,
                                            out_scores, out_boxes);
}